// RoIHeads_27204322853172
// MI455X (gfx1250) — compile-verified
//
#include <hip/hip_runtime.h>
#include <stdint.h>
#include <math.h>

// ---------------- problem constants (match reference) ----------------
#define BATCH      8
#define NPROP      4096
#define NDET       100
#define IMG_WH     512.0f
#define BBOX_CLIPV 4.135166556742356f     // log(1000/16)
#define SCORE_THR  0.05f
#define NMS_THR    0.5f
#define MIN_SIZE   1e-2f

#define NTHREADS   1024
#define PER_THREAD (NPROP / NTHREADS)     // 4
#define NWORDS     (NPROP / 32)           // 128

typedef unsigned int u32x4 __attribute__((ext_vector_type(4)));
typedef int          i32x4 __attribute__((ext_vector_type(4)));
typedef int          i32x8 __attribute__((ext_vector_type(8)));

// Low 32 bits of a generic pointer into the LDS aperture are the wave-relative
// LDS byte address (ISA 10.2).
__device__ __forceinline__ unsigned lds_addr_of(const void* p) {
    return (unsigned)(unsigned long long)p;
}

// ---------------- CDNA5 Tensor Data Mover: 1-D tile, 4B elements ------
__device__ __forceinline__ void tdm_load_1d(unsigned lds_byte_addr,
                                            const void* gsrc,
                                            unsigned nelem /* floats */) {
    unsigned long long ga = (unsigned long long)gsrc;
    u32x4 g0; i32x8 g1; i32x4 g2 = {0, 0, 0, 0}; i32x4 g3 = {0, 0, 0, 0};
    // D# group 0: count=1, lds_addr, global_addr[56:0], type=2 ("image")
    g0[0] = 1u;
    g0[1] = lds_byte_addr;
    g0[2] = (unsigned)(ga & 0xFFFFFFFFull);
    g0[3] = (unsigned)((ga >> 32) & 0x01FFFFFFull) | (2u << 30);
    // D# group 1: data_size=4B (code 2); tensor_dim0=nelem; tensor_dim1=1;
    // tile_dim0=nelem; tile_dim1=1; strides = nelem.
    g1[0] = (int)(2u << 16);
    g1[1] = (int)((nelem & 0xFFFFu) << 16);                      // dim0 lo16 -> [63:48]
    g1[2] = (int)(((nelem >> 16) & 0xFFFFu) | (1u << 16));       // dim0 hi16 | dim1 lo16
    g1[3] = (int)((nelem & 0xFFFFu) << 16);                      // dim1 hi16(0) | tile_dim0
    g1[4] = 1;                                                   // tile_dim1=1, tile_dim2=0
    g1[5] = (int)nelem;                                          // dim0_stride lo32
    g1[6] = (int)((nelem & 0xFFFFu) << 16);                      // dim0_stride hi16(0)|dim1_stride lo16
    g1[7] = (int)((nelem >> 16) & 0xFFFFu);                      // dim1_stride hi
#if defined(__clang_major__) && __clang_major__ >= 23
    i32x8 gx = {0, 0, 0, 0, 0, 0, 0, 0};
    __builtin_amdgcn_tensor_load_to_lds(g0, g1, g2, g3, gx, 0);
#else
    __builtin_amdgcn_tensor_load_to_lds(g0, g1, g2, g3, 0);
#endif
}

// ---------------- CDNA5 async global->LDS gather ----------------------
__device__ __forceinline__ void async_gather_b128(unsigned lds_dst, const void* src) {
    asm volatile("global_load_async_to_lds_b128 %0, %1, off"
                 :: "v"(lds_dst), "v"((unsigned long long)src) : "memory");
}
__device__ __forceinline__ void async_gather_b32(unsigned lds_dst, const void* src) {
    asm volatile("global_load_async_to_lds_b32 %0, %1, off"
                 :: "v"(lds_dst), "v"((unsigned long long)src) : "memory");
}
__device__ __forceinline__ void wait_async0() {
    asm volatile("s_wait_asynccnt 0" ::: "memory");
}

// ======================================================================
__global__ __launch_bounds__(NTHREADS)
void roi_postprocess_kernel(const float* __restrict__ logits,  // [B*N, 2]
                            const float* __restrict__ regs,    // [B*N, 8]
                            const float* __restrict__ props,   // [B, N, 4]
                            float* __restrict__ out,           // 4800 floats
                            float* __restrict__ ws)            // >= 640 KB
{
    // LDS: 64KB box buffer (proposals, then sorted boxes) +
    //      32KB key buffer (logits, then sort keys, then sorted scores) + tail.
    __shared__ float4             sBufA[NPROP];        // 64 KB
    __shared__ unsigned long long sKeys[NPROP];        // 32 KB
    __shared__ unsigned           sRemoved[NWORDS];
    __shared__ unsigned           sKeepMask[NWORDS];
    __shared__ int                sPrefix[NWORDS];
    __shared__ int                sCur;
    __shared__ int                sKept;

    const int tid  = threadIdx.x;
    const int img  = blockIdx.x;
    const int lane = tid & 31;
    const int wave = tid >> 5;

    float4* wsBoxes  = (float4*)ws + (size_t)img * NPROP;
    float*  wsScores = (float*)((float4*)ws + (size_t)BATCH * NPROP) + (size_t)img * NPROP;

    // ---- zero this image's output slice (harness poisons d_out) ----
    for (int t = tid; t < 6 * NDET; t += NTHREADS) {
        if (t < 4 * NDET)      out[img * 4 * NDET + t] = 0.0f;
        else if (t < 5 * NDET) out[BATCH * 4 * NDET + img * NDET + (t - 4 * NDET)] = 0.0f;
        else                   out[BATCH * 5 * NDET + img * NDET + (t - 5 * NDET)] = 0.0f;
    }
    for (int t = tid; t < NWORDS; t += NTHREADS) { sRemoved[t] = 0u; sKeepMask[t] = 0u; }
    if (tid == 0) sKept = 0;

    // ---- stage proposals + logits into LDS via the Tensor Data Mover ----
    if (tid == 0) {
        tdm_load_1d(lds_addr_of(sBufA), props  + (size_t)img * NPROP * 4, NPROP * 4);
        tdm_load_1d(lds_addr_of(sKeys), logits + (size_t)img * NPROP * 2, NPROP * 2);
        __builtin_amdgcn_s_wait_tensorcnt(0);
    }
    __syncthreads();

    const float* sProp = (const float*)sBufA;
    const float* sLog  = (const float*)sKeys;

    // ---- decode + score + clip + validity; emit rows to ws; build keys ----
    for (int s = 0; s < PER_THREAD; ++s) {
        const int i = tid + s * NTHREADS;
        const float x1 = sProp[i * 4 + 0], y1 = sProp[i * 4 + 1];
        const float x2 = sProp[i * 4 + 2], y2 = sProp[i * 4 + 3];
        const float l0 = sLog[i * 2 + 0],  l1 = sLog[i * 2 + 1];
        // class-1 deltas live in columns 4..7 of [N, 4C]
        const float4 r = *(const float4*)(regs + ((size_t)img * NPROP + i) * 8 + 4);

        const float w  = x2 - x1, h = y2 - y1;
        const float cx = x1 + 0.5f * w, cy = y1 + 0.5f * h;
        const float dx = r.x / 10.0f, dy = r.y / 10.0f;
        const float dw = fminf(r.z / 5.0f, BBOX_CLIPV);
        const float dh = fminf(r.w / 5.0f, BBOX_CLIPV);
        const float pcx = dx * w + cx, pcy = dy * h + cy;
        const float pw  = expf(dw) * w, ph = expf(dh) * h;
        float bx1 = fminf(fmaxf(pcx - 0.5f * pw, 0.0f), IMG_WH);
        float by1 = fminf(fmaxf(pcy - 0.5f * ph, 0.0f), IMG_WH);
        float bx2 = fminf(fmaxf(pcx + 0.5f * pw, 0.0f), IMG_WH);
        float by2 = fminf(fmaxf(pcy + 0.5f * ph, 0.0f), IMG_WH);

        const float m  = fmaxf(l0, l1);
        const float e0 = expf(l0 - m), e1 = expf(l1 - m);
        const float sc = e1 / (e0 + e1);

        const bool valid = (sc > SCORE_THR) && ((bx2 - bx1) >= MIN_SIZE) && ((by2 - by1) >= MIN_SIZE);
        const float scm  = valid ? sc : -INFINITY;

        wsBoxes[i]  = make_float4(bx1, by1, bx2, by2);
        wsScores[i] = scm;

        const unsigned ub = __float_as_uint(scm);
        const unsigned o  = (ub & 0x80000000u) ? ~ub : (ub | 0x80000000u); // order-preserving
        sKeys[i] = ((unsigned long long)(~o) << 32) | (unsigned)i;          // asc key == desc score
    }
    __threadfence();   // publish ws rows before the async gather reads them

    // ---- in-LDS bitonic sort of 4096 u64 keys (ascending) ----
    for (unsigned k = 2; k <= NPROP; k <<= 1) {
        for (unsigned jj = k >> 1; jj > 0; jj >>= 1) {
            __syncthreads();
            for (int s = 0; s < PER_THREAD; ++s) {
                const unsigned i = (unsigned)tid + (unsigned)s * NTHREADS;
                const unsigned ixj = i ^ jj;
                if (ixj > i) {
                    const unsigned long long a = sKeys[i], b = sKeys[ixj];
                    const bool up = ((i & k) == 0);
                    if (up ? (a > b) : (a < b)) { sKeys[i] = b; sKeys[ixj] = a; }
                }
            }
        }
    }
    __syncthreads();

    // ---- permute rows into LDS with CDNA5 async global->LDS gather ----
    unsigned myIdx[PER_THREAD];
    for (int s = 0; s < PER_THREAD; ++s)
        myIdx[s] = (unsigned)(sKeys[tid + s * NTHREADS] & 0xFFFFFFFFull);
    __syncthreads();                       // everyone captured keys; safe to overwrite

    float* sScores = (float*)sKeys;        // 16 KB of the key buffer
    for (int s = 0; s < PER_THREAD; ++s) {
        const int p = tid + s * NTHREADS;
        async_gather_b128(lds_addr_of(&sBufA[p]),    wsBoxes  + myIdx[s]);
        async_gather_b32 (lds_addr_of(&sScores[p]),  wsScores + myIdx[s]);
    }
    wait_async0();
    __syncthreads();

    // ---- greedy NMS over sorted boxes; early exit at 100 detections ----
    const float4* sBox = sBufA;
    for (int i = 0; i < NPROP; ++i) {
        if (tid == 0) {
            const bool rem = (sRemoved[i >> 5] >> (i & 31)) & 1u;
            const bool v   = sScores[i] > -INFINITY;
            const int  kp  = (v && !rem) ? 1 : 0;
            if (kp) { sKeepMask[i >> 5] |= (1u << (i & 31)); sKept++; }
            sCur = kp;
        }
        __syncthreads();
        if (sKept >= NDET) break;          // remaining boxes can never be output
        if (sCur) {
            const float4 bi = sBox[i];
            const float  ai = (bi.z - bi.x) * (bi.w - bi.y);
            for (int kk = 0; kk < 4; ++kk) {
                const int wd = wave + kk * 32;      // each wave owns 4 mask words
                const int j  = (wd << 5) + lane;
                bool sup = false;
                if (j > i) {
                    const float4 bj = sBox[j];
                    const float ix = fminf(bi.z, bj.z) - fmaxf(bi.x, bj.x);
                    const float iy = fminf(bi.w, bj.w) - fmaxf(bi.y, bj.y);
                    const float inter = fmaxf(ix, 0.0f) * fmaxf(iy, 0.0f);
                    const float aj = (bj.z - bj.x) * (bj.w - bj.y);
                    sup = (inter / (ai + aj - inter)) > NMS_THR;
                }
                const unsigned bal = (unsigned)__ballot(sup);   // wave32
                if (lane == 0 && bal) sRemoved[wd] |= bal;      // single writer / word
            }
        }
        __syncthreads();
    }
    __syncthreads();

    // ---- ranks via popcount prefix; scatter top-100 ----
    if (tid == 0) {
        int acc = 0;
        for (int wdd = 0; wdd < NWORDS; ++wdd) { sPrefix[wdd] = acc; acc += __popc(sKeepMask[wdd]); }
    }
    __syncthreads();
    for (int s = 0; s < PER_THREAD; ++s) {
        const int i = tid + s * NTHREADS;
        const unsigned mask = sKeepMask[i >> 5];
        if ((mask >> (i & 31)) & 1u) {
            const int rank = sPrefix[i >> 5] + __popc(mask & ((1u << (i & 31)) - 1u));
            if (rank < NDET) {
                const float4 bx = sBox[i];
                float* ob = out + (img * NDET + rank) * 4;
                ob[0] = bx.x; ob[1] = bx.y; ob[2] = bx.z; ob[3] = bx.w;
                out[BATCH * 4 * NDET + img * NDET + rank] = sScores[i];
                out[BATCH * 5 * NDET + img * NDET + rank] = 1.0f;   // label = 1 (only fg class)
            }
        }
    }
}

// ======================================================================
extern "C" void kernel_launch(void* const* d_in, const int* in_sizes, int n_in,
                              void* d_out, int out_size, void* d_ws, size_t ws_size,
                              hipStream_t stream) {
    (void)in_sizes; (void)n_in; (void)out_size; (void)ws_size;
    const float* logits = (const float*)d_in[0];   // [B*N, 2]
    const float* regs   = (const float*)d_in[1];   // [B*N, 8]
    const float* props  = (const float*)d_in[2];   // [B, N, 4]
    roi_postprocess_kernel<<<dim3(BATCH), dim3(NTHREADS), 0, stream>>>(
        logits, regs, props, (float*)d_out, (float*)d_ws);
}